// wootGlobalToLocal_69879117906797
// MI455X (gfx1250) — compile-verified
//
#include <hip/hip_runtime.h>
#include <hip/hip_bf16.h>
#include <stdint.h>

#define TPB  256
#define CAND 10
#define EPSF 1e-9f

#if defined(__HIP_DEVICE_COMPILE__) && defined(__gfx1250__) && __has_builtin(__builtin_amdgcn_tensor_load_to_lds)
#define USE_TDM 1
#else
#define USE_TDM 0
#endif

typedef unsigned int u32x4 __attribute__((ext_vector_type(4)));
typedef int          i32x8 __attribute__((ext_vector_type(8)));
typedef int          i32x4 __attribute__((ext_vector_type(4)));

struct V3 { float x, y, z; };

__device__ __forceinline__ V3 v3load(const float* __restrict__ p) {
    V3 r; r.x = p[0]; r.y = p[1]; r.z = p[2]; return r;
}
__device__ __forceinline__ V3 vsub(V3 a, V3 b) { return {a.x - b.x, a.y - b.y, a.z - b.z}; }
__device__ __forceinline__ float vdot(V3 a, V3 b) { return a.x * b.x + a.y * b.y + a.z * b.z; }
__device__ __forceinline__ V3 vcross(V3 a, V3 b) {
    return {a.y * b.z - a.z * b.y, a.z * b.x - a.x * b.z, a.x * b.y - a.y * b.x};
}

// Branch-free point-to-triangle "merged" distance, replicating the JAX reference
// (sum of d_i * t_i over the 7 region flags, including overlapping-flag semantics).
__device__ __forceinline__ float tri_merged(V3 q, V3 v0, V3 v1, V3 v2) {
    V3 e10 = vsub(v1, v0), e21 = vsub(v2, v1), e02 = vsub(v0, v2);
    V3 fN  = vcross(e02, e10);                 // == -cross(e10, e02)
    V3 qv0 = vsub(q, v0), qv1 = vsub(q, v1), qv2 = vsub(q, v2);

    float uab = vdot(qv0, e10) / fmaxf(vdot(e10, e10), EPSF);
    float ubc = vdot(qv1, e21) / fmaxf(vdot(e21, e21), EPSF);
    float uca = vdot(qv2, e02) / fmaxf(vdot(e02, e02), EPSF);

    bool na4 = vdot(vcross(fN, e10), qv0) <= 0.0f;
    bool na5 = vdot(vcross(fN, e21), qv1) <= 0.0f;
    bool na6 = vdot(vcross(fN, e02), qv2) <= 0.0f;

    bool t1 = (uca > 1.0f) && (uab < 0.0f);
    bool t2 = (uab > 1.0f) && (ubc < 0.0f);
    bool t3 = (ubc > 1.0f) && (uca < 0.0f);
    bool t4 = (uab >= 0.0f) && (uab <= 1.0f) && na4;
    bool t5 = (ubc >= 0.0f) && (ubc <= 1.0f) && na5 && !t4;
    bool t6 = (uca >= 0.0f) && (uca <= 1.0f) && na6 && !t4 && !t5;
    bool t0 = !(t1 | t2 | t3 | t4 | t5 | t6);

    float d0 = vdot(qv0, fN) / fmaxf(sqrtf(vdot(fN, fN)), EPSF);
    float d1 = sqrtf(vdot(qv0, qv0));
    float d2 = sqrtf(vdot(qv1, qv1));
    float d3 = sqrtf(vdot(qv2, qv2));
    V3 w4 = {qv0.x - e10.x * uab, qv0.y - e10.y * uab, qv0.z - e10.z * uab};
    V3 w5 = {qv1.x - e21.x * ubc, qv1.y - e21.y * ubc, qv1.z - e21.z * ubc};
    V3 w6 = {qv2.x - e02.x * uca, qv2.y - e02.y * uca, qv2.z - e02.z * uca};
    float d4 = sqrtf(vdot(w4, w4));
    float d5 = sqrtf(vdot(w5, w5));
    float d6 = sqrtf(vdot(w6, w6));

    return d0 * (float)t0 + d1 * (float)t1 + d2 * (float)t2 + d3 * (float)t3 +
           d4 * (float)t4 + d5 * (float)t5 + d6 * (float)t6;
}

extern "C" __global__ void __launch_bounds__(TPB)
triCandDist_gfx1250_kernel(const float* __restrict__ qxyz,
                           const float* __restrict__ verts,
                           const int*   __restrict__ faces,
                           const int*   __restrict__ cand,
                           int*         __restrict__ out_id,
                           float*       __restrict__ out_md,
                           int n) {
    __shared__ float sq[TPB * 3];
    const int base = (int)blockIdx.x * TPB;

#if USE_TDM
    // Stage the 256-point (3 KB) query tile into LDS with the Tensor Data Mover.
    // Only wave 0 issues the DMA (TDM ignores EXEC; one issue per block).
    if (threadIdx.x < 32u) {
        long long remll = (long long)n * 3 - (long long)base * 3;
        unsigned  rem   = (remll >= (long long)(TPB * 3)) ? (TPB * 3)
                                                          : (remll < 0 ? 0u : (unsigned)remll);
        unsigned long long ga = (unsigned long long)(uintptr_t)(qxyz + (size_t)base * 3);
        unsigned lds = (unsigned)(uintptr_t)(&sq[0]);

        // D# group 0: count=1 | lds_addr | global_addr[56:0] | type=2
        u32x4 g0;
        g0[0] = 1u;
        g0[1] = lds;
        g0[2] = (unsigned)(ga & 0xFFFFFFFFull);
        g0[3] = ((unsigned)((ga >> 32) & 0x01FFFFFFull)) | 0x80000000u; // type=2 at [127:126]

        // D# group 1: wg_mask=0, data_size=4B, 1-D tensor: tensor_dim0=rem,
        // tensor_dim1=1, tile_dim0=768, stride0=rem (unused for 1 row).
        i32x8 g1;
        g1[0] = (int)(2u << 16);                         // data_size = 4 bytes
        g1[1] = (int)((rem & 0xFFFFu) << 16);            // tensor_dim0[15:0] @ bits 63:48
        g1[2] = (int)((rem >> 16) | (1u << 16));         // tensor_dim0[31:16], tensor_dim1=1
        g1[3] = (int)((unsigned)(TPB * 3) << 16);        // tensor_dim1[31:16]=0, tile_dim0=768
        g1[4] = 0;                                       // tile_dim1=0, tile_dim2=0 (unused)
        g1[5] = (int)rem;                                // tensor_dim0_stride[31:0]
        g1[6] = 0;
        g1[7] = 0;

        i32x4 z4 = {0, 0, 0, 0};                         // groups 2/3 unused (<=2-D tensor)
        i32x8 z8 = {0, 0, 0, 0, 0, 0, 0, 0};             // extra group (clang-23 6-arg form)
        __builtin_amdgcn_tensor_load_to_lds(g0, g1, z4, z4, z8, 0);
        __builtin_amdgcn_s_wait_tensorcnt(0);            // data landed in LDS
    }
#else
    {
        int gi = base + (int)threadIdx.x;
        if (gi < n) {
            sq[threadIdx.x * 3 + 0] = qxyz[gi * 3 + 0];
            sq[threadIdx.x * 3 + 1] = qxyz[gi * 3 + 1];
            sq[threadIdx.x * 3 + 2] = qxyz[gi * 3 + 2];
        }
    }
#endif
    __syncthreads();

    const int gi = base + (int)threadIdx.x;
    if (gi >= n) return;

    V3 q = {sq[threadIdx.x * 3 + 0], sq[threadIdx.x * 3 + 1], sq[threadIdx.x * 3 + 2]};

    const int* __restrict__ cp = cand + (size_t)gi * CAND;

    float bestA  = INFINITY;
    float bestM  = 0.0f;
    int   bestId = 0;

    // Two batches of 5 candidates: issue 5 cand-id loads, then 5 face-index
    // triplets, then 15 vertex loads (b96 each) before any math -> deep MLP
    // against the L2 gather latency, while keeping VGPR pressure moderate.
#pragma unroll
    for (int cc = 0; cc < CAND; cc += 5) {
        int fid[5], ia[5], ib[5], ic[5];
#pragma unroll
        for (int k = 0; k < 5; ++k) fid[k] = cp[cc + k];
#pragma unroll
        for (int k = 0; k < 5; ++k) {
            const int* __restrict__ f = faces + 3 * fid[k];
            ia[k] = f[0]; ib[k] = f[1]; ic[k] = f[2];
        }
        V3 a[5], b[5], c[5];
#pragma unroll
        for (int k = 0; k < 5; ++k) {
            a[k] = v3load(verts + 3 * (size_t)ia[k]);
            b[k] = v3load(verts + 3 * (size_t)ib[k]);
            c[k] = v3load(verts + 3 * (size_t)ic[k]);
        }
#pragma unroll
        for (int k = 0; k < 5; ++k) {
            float m  = tri_merged(q, a[k], b[k], c[k]);
            float am = fabsf(m);
            if (am < bestA) {            // strict '<' == argmin first-on-tie
                bestA  = am;
                bestM  = m;
                bestId = fid[k];
            }
        }
    }

    out_id[gi] = bestId;
    out_md[gi] = bestM;
}

extern "C" void kernel_launch(void* const* d_in, const int* in_sizes, int n_in,
                              void* d_out, int out_size, void* d_ws, size_t ws_size,
                              hipStream_t stream) {
    const float* qxyz  = (const float*)d_in[0];   // (N,3)  f32
    const float* verts = (const float*)d_in[1];   // (V,3)  f32
    const int*   faces = (const int*)d_in[2];     // (F,3)  i32
    const int*   cand  = (const int*)d_in[3];     // (N,C)  i32

    const int n = in_sizes[0] / 3;

    // Tuple output, concatenated flat in return order:
    //   [0, n)   : ret (int32 face ids, written as raw int32 bits)
    //   [n, 2n)  : md  (float32 merged distances)
    int*   out_id = (int*)d_out;
    float* out_md = (float*)d_out + n;

    const int blocks = (n + TPB - 1) / TPB;
    hipLaunchKernelGGL(triCandDist_gfx1250_kernel, dim3(blocks), dim3(TPB), 0, stream,
                       qxyz, verts, faces, cand, out_id, out_md, n);
}